// ODELoraLinear_19413252177924
// MI455X (gfx1250) — compile-verified
//
#include <hip/hip_runtime.h>
#include <hip/hip_bf16.h>
#include <math.h>

// ---------------------------------------------------------------------------
// ODELoraLinear on MI455X (gfx1250), fp32.
//
// Math collapse: the Fourier transform is a fixed orthogonal matrix Q applied
// along the output dim, and the expert sum commutes with it:
//   out = x @ W_base^T + b + 0.5 * ( (x@enc0^T) @ A0^T + (x@enc1^T) @ A1^T )
//   A0 = Q @ (sum_e coefs0*mask0)   [4096 x 32]
//   A1 = Q @ (sum_e coefs1*mask1)   [4096 x 8]
// Dominant cost: one fp32 GEMM M=8192 N=4096 K=4096 (2.75e11 FLOP) on the
// fp32 matrix pipe (V_WMMA_F32_16X16X4_F32). Global->LDS staging uses the
// Tensor Data Mover (TENSOR_LOAD_TO_LDS + s_wait_tensorcnt) with D#-side LDS
// padding, double buffered -- no VGPR staging, no spills.
// ---------------------------------------------------------------------------

#ifndef __has_builtin
#define __has_builtin(x) 0
#endif
#if __has_builtin(__builtin_amdgcn_tensor_load_to_lds)
#define HAVE_TDM 1
#else
#define HAVE_TDM 0
#endif

typedef __attribute__((ext_vector_type(2))) float v2f;
typedef __attribute__((ext_vector_type(8))) float v8f;
typedef __attribute__((ext_vector_type(4))) unsigned int u32x4;
typedef __attribute__((ext_vector_type(4))) int i32x4;
typedef __attribute__((ext_vector_type(8))) int i32x8;

#define M_TOT 8192           // B*S = 4*2048
#define K_DIM 4096           // D_in
#define N_DIM 4096           // D_out
#define R0    32
#define R1    8
#define NEXP  3
#define SCALING 0.5f         // 16/32

// workspace layout, in floats
#define WS_W0SUM 0                         // 4096*32  = 131072
#define WS_W1SUM (WS_W0SUM + N_DIM * R0)   // 4096*8   =  32768
#define WS_A0    (WS_W1SUM + N_DIM * R1)   // 4096*32
#define WS_A1    (WS_A0 + N_DIM * R0)      // 4096*8
#define WS_E0    (WS_A1 + N_DIM * R1)      // 8192*32  = 262144
#define WS_E1    (WS_E0 + M_TOT * R0)      // 8192*8   =  65536
// total ~655,360 floats = 2.62 MB

// ---------------------------------------------------------------------------
// Kernel 1: masked expert sum.  W0sum[o,r] = sum_e coefs0[e,o,r]*mask0[e,o,r]
// ---------------------------------------------------------------------------
__global__ void k_masked_sum(const float* __restrict__ c0,
                             const float* __restrict__ c1,
                             const int*   __restrict__ m0,
                             const int*   __restrict__ m1,
                             float* __restrict__ w0s,
                             float* __restrict__ w1s) {
  int i = blockIdx.x * blockDim.x + threadIdx.x;
  const int n0 = N_DIM * R0, n1 = N_DIM * R1;
  if (i < n0) {
    float s = 0.f;
    for (int e = 0; e < NEXP; ++e) {
      int idx = e * n0 + i;
      s += c0[idx] * (m0[idx] ? 1.f : 0.f);
    }
    w0s[i] = s;
  } else if (i < n0 + n1) {
    int j = i - n0;
    float s = 0.f;
    for (int e = 0; e < NEXP; ++e) {
      int idx = e * n1 + j;
      s += c1[idx] * (m1[idx] ? 1.f : 0.f);
    }
    w1s[j] = s;
  }
}

// ---------------------------------------------------------------------------
// Kernel 2: A0 = Q @ W0sum, A1 = Q @ W1sum, Q built analytically per row.
// ---------------------------------------------------------------------------
__global__ void k_fourier_mix(const float* __restrict__ w0s,
                              const float* __restrict__ w1s,
                              float* __restrict__ A0,
                              float* __restrict__ A1) {
  __shared__ float qrow[K_DIM];
  const int j = blockIdx.x;
  const int t = threadIdx.x;
  const float inv_sqrt_n = 0.015625f;          // 1/sqrt(4096)
  const float s2n        = 0.022097086912079f; // sqrt(2/4096)
  const float w          = 6.28318530717958647692f / (float)K_DIM;
  for (int k = t; k < K_DIM; k += 256) {
    float q;
    if (k == 0) {
      q = inv_sqrt_n;                          // DC
    } else if (k == K_DIM - 1) {
      q = (j & 1) ? -inv_sqrt_n : inv_sqrt_n;  // Nyquist (n even)
    } else {
      int f  = (k & 1) ? ((k + 1) >> 1) : (k >> 1);
      int ph = (f * j) & (K_DIM - 1);          // exact phase mod n
      float th = w * (float)ph;
      float s, c;
      __sincosf(th, &s, &c);
      q = (k & 1) ? s2n * c : s2n * s;         // odd idx: cos, even idx: sin
    }
    qrow[k] = q;
  }
  __syncthreads();
  if (t < R0) {
    float acc = 0.f;
    for (int k = 0; k < K_DIM; ++k) acc += qrow[k] * w0s[k * R0 + t];
    A0[j * R0 + t] = acc;
  } else if (t < R0 + R1) {
    int r = t - R0;
    float acc = 0.f;
    for (int k = 0; k < K_DIM; ++k) acc += qrow[k] * w1s[k * R1 + r];
    A1[j * R1 + r] = acc;
  }
}

// ---------------------------------------------------------------------------
// Kernel 3: encoders. e0[m,r] = x[m,:] . enc0[r,:], e1 similarly.
// ---------------------------------------------------------------------------
__global__ void k_encode(const float* __restrict__ x,
                         const float* __restrict__ enc0,
                         const float* __restrict__ enc1,
                         float* __restrict__ e0,
                         float* __restrict__ e1) {
  __shared__ float xr[K_DIM];
  const int m = blockIdx.x, t = threadIdx.x;
  const float4* xv  = (const float4*)(x + (size_t)m * K_DIM);
  float4*       xrv = (float4*)xr;
  for (int i = t; i < K_DIM / 4; i += 256) xrv[i] = xv[i];
  __syncthreads();
  if (t < R0) {
    const float* er = enc0 + (size_t)t * K_DIM;
    float acc = 0.f;
    for (int k = 0; k < K_DIM; ++k) acc += xr[k] * er[k];
    e0[m * R0 + t] = acc;
  } else if (t < R0 + R1) {
    int r = t - R0;
    const float* er = enc1 + (size_t)r * K_DIM;
    float acc = 0.f;
    for (int k = 0; k < K_DIM; ++k) acc += xr[k] * er[k];
    e1[m * R1 + r] = acc;
  }
}

// ---------------------------------------------------------------------------
// Kernel 4: main fp32 GEMM. 256 threads = 8 wave32s, tile 128x128, wave tile
// 64x32 = 4x2 accumulators of v8f. K staged in 32-wide slabs.
//
// TDM staging: wave 0 issues two TENSOR_LOAD_TO_LDS (A tile, B tile) per
// slab. The D# describes a 32x128 fp32 tile of a row-major tensor with
// dim0_stride = 4096, and LDS padding of 4 DWORDs every 32 DWORDs so rows
// land at stride 36 floats (16B-aligned rows, conflict-reduced b64 fragment
// reads). Double buffered: issue slab kb+1, s_wait_tensorcnt 2 retires slab
// kb (TDM is in-order per wave), barrier publishes LDS to all waves.
//
// Fragment layouts per ISA 7.12.2 (16x4 f32 A / 4x16 f32 B):
//   lanes 0-15 : K = k0+0,+1 ; lanes 16-31 : K = k0+2,+3  -> one v2f per lane.
// C/D: VGPR v, lane L -> row M = v + 8*(L>>4), col N = L&15.
// ---------------------------------------------------------------------------
#define BM 128
#define BN 128
#define BK 32
#define LDSK 36
#define TILE_F (BM * LDSK)   // floats per LDS tile buffer

#if HAVE_TDM
// 2D TDM load: 32x128 fp32 tile, row stride K_DIM elements, +4DW pad / 32DW.
// This toolchain's builtin is the 6-arg form:
//   (u32x4 grp0, i32x8 grp1, i32x4 grp2, i32x4 grp3, i32x8 extra, i32 cpol)
// Groups 2/3 (and the extra group) are zero: 2D tile => tile_dim2/3/4 unused.
__device__ __forceinline__ void tdm_load_tile(unsigned int ldsAddr,
                                              const float* gptr,
                                              unsigned int tensor_d1) {
  unsigned long long ga = (unsigned long long)(uintptr_t)gptr;
  u32x4 g0;
  g0.x = 1u;                                  // count=1, user descriptor
  g0.y = ldsAddr;                             // LDS byte address
  g0.z = (unsigned int)ga;                    // global_addr[31:0]
  g0.w = (unsigned int)(ga >> 32) | (2u << 30); // addr[56:32] | type=2
  i32x8 g1;
  // data_size=4B (2<<16), pad_enable (1<<20),
  // pad_interval=32 DW (4<<22), pad_amount=4 DW (3<<25)
  g1[0] = (int)((2u << 16) | (1u << 20) | (4u << 22) | (3u << 25));
  g1[1] = (int)((unsigned)K_DIM << 16);       // tensor_dim0[15:0] in high half
  g1[2] = (int)(((unsigned)K_DIM >> 16) | (tensor_d1 << 16));
  g1[3] = (int)((tensor_d1 >> 16) | ((unsigned)BK << 16)); // tile_dim0=32
  g1[4] = (int)BM;                            // tile_dim1=128, tile_dim2=0
  g1[5] = (int)K_DIM;                         // tensor_dim0_stride[31:0]
  g1[6] = 0;                                  // stride0 hi | stride1 lo
  g1[7] = 0;
  i32x4 z4 = {0, 0, 0, 0};
  i32x8 z8 = {0, 0, 0, 0, 0, 0, 0, 0};
  __builtin_amdgcn_tensor_load_to_lds(g0, g1, z4, z4, z8, 0);
}
#endif

__global__ __launch_bounds__(256) void k_gemm_wmma(
    const float* __restrict__ x, const float* __restrict__ Wb,
    const float* __restrict__ bias,
    const float* __restrict__ A0, const float* __restrict__ A1,
    const float* __restrict__ e0, const float* __restrict__ e1,
    float* __restrict__ out) {
  __shared__ float As[2 * TILE_F];
  __shared__ float Bs[2 * TILE_F];

  const int t    = threadIdx.x;
  const int lane = t & 31;           // wave32
  const int wv   = t >> 5;
  const int wm   = wv & 1;           // 2 wave rows  (64 M each)
  const int wn   = wv >> 1;          // 4 wave cols  (32 N each)
  const int mBase = blockIdx.y * BM;
  const int nBase = blockIdx.x * BN;

  v8f zero = {0.f, 0.f, 0.f, 0.f, 0.f, 0.f, 0.f, 0.f};
  v8f acc[4][2];
#pragma unroll
  for (int mt = 0; mt < 4; ++mt)
#pragma unroll
    for (int nt = 0; nt < 2; ++nt) acc[mt][nt] = zero;

  const int NKB = K_DIM / BK;

#if HAVE_TDM
  // ---- Tensor Data Mover double-buffered pipeline -------------------------
  const unsigned int asLds0 = (unsigned int)(uintptr_t)&As[0];
  const unsigned int bsLds0 = (unsigned int)(uintptr_t)&Bs[0];
  if (wv == 0) {
    tdm_load_tile(asLds0, x  + (size_t)mBase * K_DIM, (unsigned)(M_TOT - mBase));
    tdm_load_tile(bsLds0, Wb + (size_t)nBase * K_DIM, (unsigned)(N_DIM - nBase));
  }
  for (int kb = 0; kb < NKB; ++kb) {
    const int buf = kb & 1;
    if (kb + 1 < NKB) {
      if (wv == 0) {
        tdm_load_tile(asLds0 + (unsigned)((buf ^ 1) * TILE_F * 4),
                      x + (size_t)mBase * K_DIM + (kb + 1) * BK,
                      (unsigned)(M_TOT - mBase));
        tdm_load_tile(bsLds0 + (unsigned)((buf ^ 1) * TILE_F * 4),
                      Wb + (size_t)nBase * K_DIM + (kb + 1) * BK,
                      (unsigned)(N_DIM - nBase));
      }
      __builtin_amdgcn_s_wait_tensorcnt(2);   // slab kb complete (in-order)
    } else {
      __builtin_amdgcn_s_wait_tensorcnt(0);
    }
    __syncthreads();                          // publish LDS to all waves
    const float* Ab = As + buf * TILE_F;
    const float* Bb = Bs + buf * TILE_F;
#pragma unroll
    for (int ks = 0; ks < BK / 4; ++ks) {
      const int k0 = ks * 4 + 2 * (lane >> 4);
      v2f aF[4], bF[2];
#pragma unroll
      for (int mt = 0; mt < 4; ++mt) {
        int row = wm * 64 + mt * 16 + (lane & 15);
        aF[mt] = *(const v2f*)(Ab + row * LDSK + k0);
      }
#pragma unroll
      for (int nt = 0; nt < 2; ++nt) {
        int col = wn * 32 + nt * 16 + (lane & 15);
        bF[nt] = *(const v2f*)(Bb + col * LDSK + k0);
      }
#pragma unroll
      for (int mt = 0; mt < 4; ++mt)
#pragma unroll
        for (int nt = 0; nt < 2; ++nt)
          acc[mt][nt] = __builtin_amdgcn_wmma_f32_16x16x4_f32(
              false, aF[mt], false, bF[nt], (short)0, acc[mt][nt], false, false);
    }
    __syncthreads();                          // all reads done before reuse
  }
#else
  // ---- fallback: register-staged single-buffer pipeline -------------------
  float4 aR[4], bR[4];
  auto loadTile = [&](int kb) {
#pragma unroll
    for (int p = 0; p < 4; ++p) {
      int idx = p * 1024 + t * 4;
      int mm = idx >> 5, kk = idx & 31;
      aR[p] = *(const float4*)(x  + (size_t)(mBase + mm) * K_DIM + kb * BK + kk);
      bR[p] = *(const float4*)(Wb + (size_t)(nBase + mm) * K_DIM + kb * BK + kk);
    }
  };
  auto storeTile = [&]() {
#pragma unroll
    for (int p = 0; p < 4; ++p) {
      int idx = p * 1024 + t * 4;
      int mm = idx >> 5, kk = idx & 31;
      *(float4*)(As + mm * LDSK + kk) = aR[p];
      *(float4*)(Bs + mm * LDSK + kk) = bR[p];
    }
  };
  loadTile(0);
  for (int kb = 0; kb < NKB; ++kb) {
    __syncthreads();
    storeTile();
    __syncthreads();
    if (kb + 1 < NKB) loadTile(kb + 1);
#pragma unroll
    for (int ks = 0; ks < BK / 4; ++ks) {
      const int k0 = ks * 4 + 2 * (lane >> 4);
      v2f aF[4], bF[2];
#pragma unroll
      for (int mt = 0; mt < 4; ++mt) {
        int row = wm * 64 + mt * 16 + (lane & 15);
        aF[mt] = *(const v2f*)(As + row * LDSK + k0);
      }
#pragma unroll
      for (int nt = 0; nt < 2; ++nt) {
        int col = wn * 32 + nt * 16 + (lane & 15);
        bF[nt] = *(const v2f*)(Bs + col * LDSK + k0);
      }
#pragma unroll
      for (int mt = 0; mt < 4; ++mt)
#pragma unroll
        for (int nt = 0; nt < 2; ++nt)
          acc[mt][nt] = __builtin_amdgcn_wmma_f32_16x16x4_f32(
              false, aF[mt], false, bF[nt], (short)0, acc[mt][nt], false, false);
    }
  }
#endif

  // ---- epilogue: bias + rank-40 LoRA update -------------------------------
#pragma unroll
  for (int nt = 0; nt < 2; ++nt) {
    const int n = nBase + wn * 32 + nt * 16 + (lane & 15);
    const float bia = bias[n];
    float a0r[R0], a1r[R1];
    {
      const float4* a0v = (const float4*)(A0 + (size_t)n * R0);
#pragma unroll
      for (int q = 0; q < R0 / 4; ++q) {
        float4 v4 = a0v[q];
        a0r[4 * q + 0] = v4.x; a0r[4 * q + 1] = v4.y;
        a0r[4 * q + 2] = v4.z; a0r[4 * q + 3] = v4.w;
      }
      const float4* a1v = (const float4*)(A1 + (size_t)n * R1);
#pragma unroll
      for (int q = 0; q < R1 / 4; ++q) {
        float4 v4 = a1v[q];
        a1r[4 * q + 0] = v4.x; a1r[4 * q + 1] = v4.y;
        a1r[4 * q + 2] = v4.z; a1r[4 * q + 3] = v4.w;
      }
    }
#pragma unroll
    for (int mt = 0; mt < 4; ++mt) {
#pragma unroll
      for (int v = 0; v < 8; ++v) {
        const int m = mBase + wm * 64 + mt * 16 + v + 8 * (lane >> 4);
        const float* er0 = e0 + (size_t)m * R0;
        const float* er1 = e1 + (size_t)m * R1;
        float upd = 0.f;
#pragma unroll
        for (int r = 0; r < R0; ++r) upd += er0[r] * a0r[r];
#pragma unroll
        for (int r = 0; r < R1; ++r) upd += er1[r] * a1r[r];
        out[(size_t)m * N_DIM + n] = acc[mt][nt][v] + bia + SCALING * upd;
      }
    }
  }
}

// ---------------------------------------------------------------------------
// Host launcher
// inputs: x, W_base, b_base, enc0, enc1, coefs0, coefs1, mask0, mask1
// ---------------------------------------------------------------------------
extern "C" void kernel_launch(void* const* d_in, const int* in_sizes, int n_in,
                              void* d_out, int out_size, void* d_ws, size_t ws_size,
                              hipStream_t stream) {
  const float* x    = (const float*)d_in[0];
  const float* Wb   = (const float*)d_in[1];
  const float* bias = (const float*)d_in[2];
  const float* enc0 = (const float*)d_in[3];
  const float* enc1 = (const float*)d_in[4];
  const float* c0   = (const float*)d_in[5];
  const float* c1   = (const float*)d_in[6];
  const int*   m0   = (const int*)d_in[7];   // bool masks as integer
  const int*   m1   = (const int*)d_in[8];

  float* ws  = (float*)d_ws;
  float* W0s = ws + WS_W0SUM;
  float* W1s = ws + WS_W1SUM;
  float* A0  = ws + WS_A0;
  float* A1  = ws + WS_A1;
  float* e0  = ws + WS_E0;
  float* e1  = ws + WS_E1;
  float* out = (float*)d_out;

  const int nsum = N_DIM * R0 + N_DIM * R1;
  k_masked_sum<<<(nsum + 255) / 256, 256, 0, stream>>>(c0, c1, m0, m1, W0s, W1s);
  k_fourier_mix<<<N_DIM, 256, 0, stream>>>(W0s, W1s, A0, A1);
  k_encode<<<M_TOT, 256, 0, stream>>>(x, enc0, enc1, e0, e1);

  dim3 grid(N_DIM / BN, M_TOT / BM);
  k_gemm_wmma<<<grid, 256, 0, stream>>>(x, Wb, bias, A0, A1, e0, e1, out);
}